// ScanAttention_66288525246583
// MI455X (gfx1250) — compile-verified
//
#include <hip/hip_runtime.h>
#include <math.h>

typedef float v2f __attribute__((ext_vector_type(2)));
typedef float v8f __attribute__((ext_vector_type(8)));

#define S_LEN 1024
#define DIMV  512
#define NH    8
#define HDV   64
#define EPSV  1e-5f

// D = A(16x4 f32) * B(4x16 f32) + C(16x16 f32), wave32 fp32 WMMA.
__device__ __forceinline__ v8f wmma_f32(v2f a, v2f b, v8f c) {
  return __builtin_amdgcn_wmma_f32_16x16x4_f32(
      /*neg_a=*/false, a, /*neg_b=*/false, b,
      /*c_mod=*/(short)0, c, /*reuse_a=*/false, /*reuse_b=*/false);
}

// ---------------------------------------------------------------------------
// Kernel 1: fused Q/K/V projections.  [B*S,512] @ [512,512] (+bias), K scaled
// by 1/8, written to per-head [b,h,s,d] layout.  One wave owns a 64x64 output
// block: 16 accumulators, A/B fragments each reused 4x -> 16 WMMA per k-step.
// ---------------------------------------------------------------------------
__global__ __launch_bounds__(32) void k_proj(
    const float* __restrict__ x,
    const float* __restrict__ wq, const float* __restrict__ bq,
    const float* __restrict__ wk, const float* __restrict__ bk,
    const float* __restrict__ wv, const float* __restrict__ bv,
    float* __restrict__ Qh, float* __restrict__ Kt, float* __restrict__ Vt) {
  const int lane = threadIdx.x;
  const int m  = lane & 15;
  const int hi = lane >> 4;
  const int row0  = blockIdx.x * 64;   // B*S = 2048 -> 32 blocks
  const int col0  = blockIdx.y * 64;   // DIM = 512  -> 8 blocks
  const int which = blockIdx.z;        // 0=Q 1=K 2=V

  const float* w    = (which == 0) ? wq : (which == 1) ? wk : wv;
  const float* bias = (which == 0) ? bq : (which == 1) ? bk : bv;
  const float scale = (which == 1) ? 0.125f : 1.0f;   // K /= sqrt(64)

  v8f acc[4][4] = {};
  const float* ar[4];
  #pragma unroll
  for (int rt = 0; rt < 4; ++rt) ar[rt] = x + (size_t)(row0 + 16 * rt + m) * DIMV;

  #pragma unroll 1
  for (int k0 = 0; k0 < DIMV; k0 += 4) {
    const int kk = k0 + 2 * hi;
    v2f a[4];
    #pragma unroll
    for (int rt = 0; rt < 4; ++rt) a[rt] = *(const v2f*)(ar[rt] + kk);
    const float* wr = w + (size_t)kk * DIMV + col0 + m;
    v2f b[4];
    #pragma unroll
    for (int ct = 0; ct < 4; ++ct) { b[ct].x = wr[16 * ct]; b[ct].y = wr[DIMV + 16 * ct]; }
    #pragma unroll
    for (int rt = 0; rt < 4; ++rt)
      #pragma unroll
      for (int ct = 0; ct < 4; ++ct)
        acc[rt][ct] = wmma_f32(a[rt], b[ct], acc[rt][ct]);
  }

  float* dst = (which == 0) ? Qh : (which == 1) ? Kt : Vt;
  #pragma unroll
  for (int ct = 0; ct < 4; ++ct) {
    const int ncol = col0 + 16 * ct + m;
    const int h = ncol >> 6, d = ncol & 63;
    const float bb = bias[ncol];
    #pragma unroll
    for (int rt = 0; rt < 4; ++rt)
      #pragma unroll
      for (int vi = 0; vi < 8; ++vi) {
        const int r = row0 + 16 * rt + vi + 8 * hi;
        const int batch = r >> 10, s = r & 1023;
        dst[(((size_t)(batch * NH + h) * S_LEN) + s) * HDV + d] =
            (acc[rt][ct][vi] + bb) * scale;
      }
  }
}

// ---------------------------------------------------------------------------
// Kernel 2: causal conv as triangular Toeplitz GEMM:  Y = T_h @ X,
// T_h[t,t'] = f_h[t-t'] (t'<=t), tiles generated on the fly from
// spectral_basis.  One wave owns 64 rows x 64 cols; the d0>=0 guard makes the
// single j-loop correct for all four row-tiles (upper triangle -> zeros).
// ---------------------------------------------------------------------------
__global__ __launch_bounds__(32) void k_conv(
    const float* __restrict__ sb,    // [S][H]
    const float* __restrict__ Kt, const float* __restrict__ Vt,
    float* __restrict__ Kc, float* __restrict__ Vc) {
  const int lane = threadIdx.x;
  const int m = lane & 15, hi = lane >> 4;
  const int ib = blockIdx.x;   // 64-row group, 0..15
  const int bh = blockIdx.y;   // 0..15
  const int h  = bh & 7;
  const int which = blockIdx.z;
  const float* src = ((which == 0) ? Kt : Vt) + (size_t)bh * S_LEN * HDV;
  float*       dst = ((which == 0) ? Kc : Vc) + (size_t)bh * S_LEN * HDV;

  v8f acc[4][4] = {};
  const int jmax = 4 * ib + 3;
  #pragma unroll 1
  for (int j = 0; j <= jmax; ++j) {
    #pragma unroll
    for (int st = 0; st < 4; ++st) {
      const int kb = 4 * st;
      v2f a[4];
      #pragma unroll
      for (int rt = 0; rt < 4; ++rt) {
        const int d0 = 16 * (4 * ib + rt - j) + m - 2 * hi - kb;
        a[rt].x = (d0     >= 0) ? sb[d0 * NH + h]       : 0.0f;
        a[rt].y = (d0 - 1 >= 0) ? sb[(d0 - 1) * NH + h] : 0.0f;
      }
      const float* s0 = src + (size_t)(16 * j + kb + 2 * hi) * HDV + m;
      v2f b[4];
      #pragma unroll
      for (int ct = 0; ct < 4; ++ct) { b[ct].x = s0[16 * ct]; b[ct].y = s0[HDV + 16 * ct]; }
      #pragma unroll
      for (int rt = 0; rt < 4; ++rt)
        #pragma unroll
        for (int ct = 0; ct < 4; ++ct)
          acc[rt][ct] = wmma_f32(a[rt], b[ct], acc[rt][ct]);
    }
  }
  #pragma unroll
  for (int rt = 0; rt < 4; ++rt)
    #pragma unroll
    for (int vi = 0; vi < 8; ++vi) {
      const size_t r = 64 * ib + 16 * rt + vi + 8 * hi;
      #pragma unroll
      for (int ct = 0; ct < 4; ++ct)
        dst[r * HDV + 16 * ct + m] = acc[rt][ct][vi];
    }
}

// ---------------------------------------------------------------------------
// Kernel 3: gates g[s] = relu(v_s^T G k_s)^2 + eps, and cum_g via LDS scan.
// One 1024-thread block per (b,h); G staged in LDS.
// ---------------------------------------------------------------------------
__global__ __launch_bounds__(1024) void k_gates(
    const float* __restrict__ Kc, const float* __restrict__ Vc,
    const float* __restrict__ gw, const float* __restrict__ gb,
    float* __restrict__ g, float* __restrict__ cumg) {
  __shared__ float Gs[HDV * HDV];
  __shared__ float sc[S_LEN];
  const int tid = threadIdx.x;
  const int bh  = blockIdx.x;
  for (int t = tid; t < HDV * HDV; t += 1024) Gs[t] = gw[t];
  __syncthreads();

  const float* krow = Kc + ((size_t)bh * S_LEN + tid) * HDV;
  const float* vrow = Vc + ((size_t)bh * S_LEN + tid) * HDV;
  float kr[HDV];
  #pragma unroll
  for (int e = 0; e < HDV; ++e) kr[e] = krow[e];
  float gl = gb[0];
  for (int d = 0; d < HDV; ++d) {
    const float vd = vrow[d];
    float acc = 0.0f;
    #pragma unroll
    for (int e = 0; e < HDV; ++e) acc += Gs[d * HDV + e] * kr[e];
    gl += vd * acc;
  }
  float gt = fmaxf(gl, 0.0f);
  gt = gt * gt + EPSV;

  sc[tid] = gt;
  __syncthreads();
  for (int off = 1; off < S_LEN; off <<= 1) {   // Hillis-Steele inclusive scan
    const float add = (tid >= off) ? sc[tid - off] : 0.0f;
    __syncthreads();
    sc[tid] += add;
    __syncthreads();
  }
  g[(size_t)bh * S_LEN + tid]    = gt;
  cumg[(size_t)bh * S_LEN + tid] = sc[tid];
}

// ---------------------------------------------------------------------------
// Kernel 4: gated causal linear attention.
// out[s,e] = (1/(cum_g[s]+eps)) * sum_{t<=s} g_t * (q_s . v_t) * k_t[e],
// then L2-normalize each 64-vector.  One wave owns 32 rows x 64 cols; the
// score tiles bounce through LDS to convert C-fragment -> A-fragment layout.
// ---------------------------------------------------------------------------
__global__ __launch_bounds__(32) void k_attn(
    const float* __restrict__ Qh, const float* __restrict__ Kc,
    const float* __restrict__ Vc, const float* __restrict__ g,
    const float* __restrict__ cumg, float* __restrict__ O) {
  __shared__ float tile[32 * 16];
  const int lane = threadIdx.x;
  const int m = lane & 15, hi = lane >> 4;
  const int ib = blockIdx.x;     // 32-row group, 0..31
  const int bh = blockIdx.y;     // 0..15
  const int batch = bh >> 3, h = bh & 7;
  const float* Q  = Qh + (size_t)bh * S_LEN * HDV;
  const float* K  = Kc + (size_t)bh * S_LEN * HDV;
  const float* V  = Vc + (size_t)bh * S_LEN * HDV;
  const float* gv = g  + (size_t)bh * S_LEN;

  const float* qrow[2];
  #pragma unroll
  for (int rt = 0; rt < 2; ++rt)
    qrow[rt] = Q + (size_t)(32 * ib + 16 * rt + m) * HDV;

  v8f acc[2][4] = {};
  const int jmax = 2 * ib + 1;
  #pragma unroll 1
  for (int j = 0; j <= jmax; ++j) {
    // ---- score tiles [2 x 16x16]: Q_tile @ Vc_tile^T over d=64
    v8f sf[2] = {};
    const float* vrow = V + (size_t)(16 * j + m) * HDV;   // B[k][n]=Vc[16j+n][k]
    #pragma unroll 4
    for (int st = 0; st < 16; ++st) {
      const int kb = 4 * st + 2 * hi;
      v2f b = *(const v2f*)(vrow + kb);
      #pragma unroll
      for (int rt = 0; rt < 2; ++rt) {
        v2f a = *(const v2f*)(qrow[rt] + kb);
        sf[rt] = wmma_f32(a, b, sf[rt]);
      }
    }
    // ---- causal mask + gate column scaling, spill to LDS in (row,col) form
    const float gcol = gv[16 * j + m];
    #pragma unroll
    for (int rt = 0; rt < 2; ++rt)
      #pragma unroll
      for (int vi = 0; vi < 8; ++vi) {
        const int r = 32 * ib + 16 * rt + vi + 8 * hi;  // query position
        const int t = 16 * j + m;                       // key position
        tile[(16 * rt + vi + 8 * hi) * 16 + m] = (r >= t) ? sf[rt][vi] * gcol : 0.0f;
      }
    __syncthreads();
    // ---- out += scores @ Kc_tile   (K-dim = 16 -> 4 WMMA k-steps)
    #pragma unroll
    for (int st = 0; st < 4; ++st) {
      const int kb = 4 * st + 2 * hi;
      const float* kp = K + (size_t)(16 * j + kb) * HDV + m;
      v2f b[4];
      #pragma unroll
      for (int ct = 0; ct < 4; ++ct) { b[ct].x = kp[16 * ct]; b[ct].y = kp[HDV + 16 * ct]; }
      #pragma unroll
      for (int rt = 0; rt < 2; ++rt) {
        v2f a;
        a.x = tile[(16 * rt + m) * 16 + kb];
        a.y = tile[(16 * rt + m) * 16 + kb + 1];
        #pragma unroll
        for (int ct = 0; ct < 4; ++ct)
          acc[rt][ct] = wmma_f32(a, b[ct], acc[rt][ct]);
      }
    }
    __syncthreads();
  }
  // ---- finalize: /(cum_g+eps), L2-normalize rows (shuffle reduce over the
  // 16 lanes of the half-wave holding each row), scatter to [b,s,h,e].
  #pragma unroll
  for (int rt = 0; rt < 2; ++rt)
    #pragma unroll
    for (int vi = 0; vi < 8; ++vi) {
      const int r = 32 * ib + 16 * rt + vi + 8 * hi;
      const float scale = 1.0f / (cumg[(size_t)bh * S_LEN + r] + EPSV);
      float e[4];
      #pragma unroll
      for (int ct = 0; ct < 4; ++ct) e[ct] = acc[rt][ct][vi] * scale;
      float p = e[0] * e[0] + e[1] * e[1] + e[2] * e[2] + e[3] * e[3];
      p += __shfl_xor(p, 1, 32);
      p += __shfl_xor(p, 2, 32);
      p += __shfl_xor(p, 4, 32);
      p += __shfl_xor(p, 8, 32);
      const float inv = 1.0f / fmaxf(sqrtf(p), EPSV);
      float* op = O + ((size_t)(batch * S_LEN + r) * DIMV) + h * HDV + m;
      #pragma unroll
      for (int ct = 0; ct < 4; ++ct) op[16 * ct] = e[ct] * inv;
    }
}

// ---------------------------------------------------------------------------
// Kernel 5: output projection  out = O @ wo + bo,  [2048,512]@[512,512].
// Same 64x64 register blocking as k_proj.
// ---------------------------------------------------------------------------
__global__ __launch_bounds__(32) void k_oproj(
    const float* __restrict__ O, const float* __restrict__ wo,
    const float* __restrict__ bo, float* __restrict__ out) {
  const int lane = threadIdx.x;
  const int m = lane & 15, hi = lane >> 4;
  const int row0 = blockIdx.x * 64, col0 = blockIdx.y * 64;

  v8f acc[4][4] = {};
  const float* ar[4];
  #pragma unroll
  for (int rt = 0; rt < 4; ++rt) ar[rt] = O + (size_t)(row0 + 16 * rt + m) * DIMV;

  #pragma unroll 1
  for (int k0 = 0; k0 < DIMV; k0 += 4) {
    const int kk = k0 + 2 * hi;
    v2f a[4];
    #pragma unroll
    for (int rt = 0; rt < 4; ++rt) a[rt] = *(const v2f*)(ar[rt] + kk);
    const float* wr = wo + (size_t)kk * DIMV + col0 + m;
    v2f b[4];
    #pragma unroll
    for (int ct = 0; ct < 4; ++ct) { b[ct].x = wr[16 * ct]; b[ct].y = wr[DIMV + 16 * ct]; }
    #pragma unroll
    for (int rt = 0; rt < 4; ++rt)
      #pragma unroll
      for (int ct = 0; ct < 4; ++ct)
        acc[rt][ct] = wmma_f32(a[rt], b[ct], acc[rt][ct]);
  }

  #pragma unroll
  for (int ct = 0; ct < 4; ++ct) {
    const float bias = bo[col0 + 16 * ct + m];
    #pragma unroll
    for (int rt = 0; rt < 4; ++rt)
      #pragma unroll
      for (int vi = 0; vi < 8; ++vi) {
        const int r = row0 + 16 * rt + vi + 8 * hi;
        out[(size_t)r * DIMV + col0 + 16 * ct + m] = acc[rt][ct][vi] + bias;
      }
  }
}

// ---------------------------------------------------------------------------
extern "C" void kernel_launch(void* const* d_in, const int* in_sizes, int n_in,
                              void* d_out, int out_size, void* d_ws, size_t ws_size,
                              hipStream_t stream) {
  const float* x  = (const float*)d_in[0];
  const float* sb = (const float*)d_in[1];
  const float* wq = (const float*)d_in[2];
  const float* bq = (const float*)d_in[3];
  const float* wk = (const float*)d_in[4];
  const float* bk = (const float*)d_in[5];
  const float* wv = (const float*)d_in[6];
  const float* bv = (const float*)d_in[7];
  const float* wo = (const float*)d_in[8];
  const float* bo = (const float*)d_in[9];
  const float* gw = (const float*)d_in[10];
  const float* gb = (const float*)d_in[11];
  float* out = (float*)d_out;

  float* ws = (float*)d_ws;
  const size_t PH = (size_t)2 * NH * S_LEN * HDV;   // 1,048,576 floats per tensor
  float* Qh = ws;            // [b,h,s,d] Q projection
  float* Kt = Qh + PH;       // pre-conv K (scaled)
  float* Vt = Kt + PH;       // pre-conv V
  float* Kc = Vt + PH;       // post-conv K
  float* Vc = Kc + PH;       // post-conv V
  float* gg = Vc + PH;       // gates      [b,h,s]
  float* cg = gg + (size_t)2 * NH * S_LEN;   // cum gates [b,h,s]
  float* O  = cg + (size_t)2 * NH * S_LEN;   // normalized attn out [b,s,dim]
  (void)ws_size; (void)in_sizes; (void)n_in; (void)out_size;

  k_proj <<<dim3(32, 8, 3), 32, 0, stream>>>(x, wq, bq, wk, bk, wv, bv, Qh, Kt, Vt);
  k_conv <<<dim3(16, 16, 2), 32, 0, stream>>>(sb, Kt, Vt, Kc, Vc);
  k_gates<<<dim3(16), 1024, 0, stream>>>(Kc, Vc, gw, gb, gg, cg);
  k_attn <<<dim3(32, 16), 32, 0, stream>>>(Qh, Kc, Vc, gg, cg, O);
  k_oproj<<<dim3(32, 8), 32, 0, stream>>>(O, wo, bo, out);
}